// DecoderBlock_fixedA_12051678232916
// MI455X (gfx1250) — compile-verified
//
#include <hip/hip_runtime.h>
#include <hip/hip_bf16.h>
#include <cstddef>
#include <cstdint>

typedef float v2f __attribute__((ext_vector_type(2)));
typedef float v8f __attribute__((ext_vector_type(8)));

static constexpr int NN = 50000;   // nodes (== 3125 * 16, tile-aligned)
static constexpr int EE = 800000;  // edges (self loops handled analytically)

// ---------------------------------------------------------------------------
// WMMA GEMM: Y[NN,K] = X[NN,DIN] * W[DIN,K], all row-major f32.
// One wave32 computes one 16x16 tile of Y using V_WMMA_F32_16X16X4_F32.
// A (16x4 f32): lane L holds A[M=L&15, K=2*(L>>4)+v] in VGPR v.
// B (4x16 f32): lane L holds B[K=2*(L>>4)+v, N=L&15] in VGPR v.
// C (16x16 f32): VGPR p, lanes 0-15 -> M=p, lanes 16-31 -> M=p+8, N=lane&15.
// ---------------------------------------------------------------------------
template <int DIN>
__global__ __launch_bounds__(32)
void gemm_xT_wmma(const float* __restrict__ X, const float* __restrict__ W,
                  float* __restrict__ Y, int K) {
  const int lane = threadIdx.x & 31;
  const int half = lane >> 4;   // 0 or 1
  const int l16  = lane & 15;
  const int row  = blockIdx.x * 16 + l16;   // A row
  const int col  = blockIdx.y * 16 + l16;   // B / C column
  v8f c = {};
#pragma unroll
  for (int k0 = 0; k0 < DIN; k0 += 4) {
    const int ka = k0 + half * 2;
    v2f a, b;
    a.x = X[(size_t)row * DIN + ka];
    a.y = X[(size_t)row * DIN + ka + 1];
    b.x = W[(size_t)ka * K + col];
    b.y = W[(size_t)(ka + 1) * K + col];
    // (neg_a, A, neg_b, B, c_mod, C, reuse_a, reuse_b)
    c = __builtin_amdgcn_wmma_f32_16x16x4_f32(false, a, false, b,
                                              (short)0, c, false, false);
  }
  const int rbase = blockIdx.x * 16 + half * 8;
#pragma unroll
  for (int p = 0; p < 8; ++p)
    Y[(size_t)(rbase + p) * K + col] = c[p];
}

// P[i, h] = sum_d X[i,d] * U[d,h], h = 0..3 (attention projections per node)
__global__ __launch_bounds__(256)
void xu_kernel(const float* __restrict__ X, const float* __restrict__ U,
               float* __restrict__ P, int din) {
  const int i = blockIdx.x * blockDim.x + threadIdx.x;
  if (i >= NN) return;
  const float* x = X + (size_t)i * din;
  float a0 = 0.f, a1 = 0.f, a2 = 0.f, a3 = 0.f;
  for (int d = 0; d < din; ++d) {
    const float xv = x[d];
    a0 += xv * U[d * 4 + 0];
    a1 += xv * U[d * 4 + 1];
    a2 += xv * U[d * 4 + 2];
    a3 += xv * U[d * 4 + 3];
  }
  P[i * 4 + 0] = a0; P[i * 4 + 1] = a1;
  P[i * 4 + 2] = a2; P[i * 4 + 3] = a3;
}

// in-degree over the E real edges (self loop added as +1 later)
__global__ __launch_bounds__(256)
void deg_count_kernel(const long long* __restrict__ dst, unsigned* __restrict__ deg) {
  const int e = blockIdx.x * blockDim.x + threadIdx.x;
  if (e < EE) atomicAdd(&deg[(int)dst[e]], 1u);
}

// Per-edge: q = softmax_h(P[s]-P[d]+c); agg[d] += sum_h q_h * Y[s][h*dout : ...]
__global__ __launch_bounds__(256)
void feast_edge_kernel(const long long* __restrict__ src,
                       const long long* __restrict__ dst,
                       const float* __restrict__ P,
                       const float* __restrict__ cb,
                       const float* __restrict__ Y,
                       float* __restrict__ agg, int dout) {
  const int e = blockIdx.x * blockDim.x + threadIdx.x;
  if (e >= EE) return;
  const int s = (int)src[e];
  const int d = (int)dst[e];
  const float l0 = P[s * 4 + 0] - P[d * 4 + 0] + cb[0];
  const float l1 = P[s * 4 + 1] - P[d * 4 + 1] + cb[1];
  const float l2 = P[s * 4 + 2] - P[d * 4 + 2] + cb[2];
  const float l3 = P[s * 4 + 3] - P[d * 4 + 3] + cb[3];
  const float mx = fmaxf(fmaxf(l0, l1), fmaxf(l2, l3));
  const float e0 = __expf(l0 - mx), e1 = __expf(l1 - mx);
  const float e2 = __expf(l2 - mx), e3 = __expf(l3 - mx);
  const float inv = 1.0f / (e0 + e1 + e2 + e3);
  const float q0 = e0 * inv, q1 = e1 * inv, q2 = e2 * inv, q3 = e3 * inv;
  const float* y = Y + (size_t)s * (4 * dout);
  float* ag = agg + (size_t)d * dout;
  for (int o = 0; o < dout; o += 4) {
    const float4 y0 = *(const float4*)(y + o);
    const float4 y1 = *(const float4*)(y + dout + o);
    const float4 y2 = *(const float4*)(y + 2 * dout + o);
    const float4 y3 = *(const float4*)(y + 3 * dout + o);
    atomicAdd(&ag[o + 0], q0 * y0.x + q1 * y1.x + q2 * y2.x + q3 * y3.x);
    atomicAdd(&ag[o + 1], q0 * y0.y + q1 * y1.y + q2 * y2.y + q3 * y3.y);
    atomicAdd(&ag[o + 2], q0 * y0.z + q1 * y1.z + q2 * y2.z + q3 * y3.z);
    atomicAdd(&ag[o + 3], q0 * y0.w + q1 * y1.w + q2 * y2.w + q3 * y3.w);
  }
}

// out[i,ch] = (agg[i,ch] + selfmsg[i,ch]) / (deg[i]+1) + bias[ch];
// optionally accumulate per-channel sum / sumsq for BatchNorm.
// stride is a multiple of dout so each thread owns one fixed channel.
template <bool WITH_BN>
__global__ __launch_bounds__(256)
void feast_finalize_kernel(const float* __restrict__ agg,
                           const float* __restrict__ Y,
                           const unsigned* __restrict__ deg,
                           const float* __restrict__ bias,
                           const float* __restrict__ cb,
                           float* __restrict__ Out,
                           float* __restrict__ sums,
                           float* __restrict__ sumsq, int dout) {
  // self-loop attention weights: softmax(c)
  const float c0 = cb[0], c1 = cb[1], c2 = cb[2], c3 = cb[3];
  const float mx = fmaxf(fmaxf(c0, c1), fmaxf(c2, c3));
  const float e0 = __expf(c0 - mx), e1 = __expf(c1 - mx);
  const float e2 = __expf(c2 - mx), e3 = __expf(c3 - mx);
  const float inv = 1.0f / (e0 + e1 + e2 + e3);
  const float q0 = e0 * inv, q1 = e1 * inv, q2 = e2 * inv, q3 = e3 * inv;

  const int total  = NN * dout;
  const int stride = gridDim.x * blockDim.x;   // 256 | blockDim -> stride % dout == 0
  int idx = blockIdx.x * blockDim.x + threadIdx.x;
  const int ch = idx % dout;                    // constant across strides
  const float b = bias[ch];
  float ls = 0.f, ls2 = 0.f;
  for (; idx < total; idx += stride) {
    const int i = idx / dout;
    const float* y = Y + (size_t)i * (4 * dout) + ch;
    const float self = q0 * y[0] + q1 * y[dout] + q2 * y[2 * dout] + q3 * y[3 * dout];
    const float t = (agg[idx] + self) / (float)(deg[i] + 1u) + b;
    Out[idx] = t;
    if (WITH_BN) { ls += t; ls2 += t * t; }
  }
  if (WITH_BN) {
    atomicAdd(&sums[ch], ls);
    atomicAdd(&sumsq[ch], ls2);
  }
}

__global__ void bn_stats_kernel(const float* __restrict__ sums,
                                const float* __restrict__ sumsq,
                                const float* __restrict__ g,
                                const float* __restrict__ beta,
                                float* __restrict__ scale,
                                float* __restrict__ shift, int dout) {
  const int ch = threadIdx.x;
  if (ch >= dout) return;
  const float m   = sums[ch] / (float)NN;
  const float var = sumsq[ch] / (float)NN - m * m;   // biased variance
  const float sc  = g[ch] * rsqrtf(var + 1e-5f);
  scale[ch] = sc;
  shift[ch] = beta[ch] - m * sc;
}

__global__ __launch_bounds__(256)
void bn_relu_kernel(float* __restrict__ X, const float* __restrict__ scale,
                    const float* __restrict__ shift, int dout) {
  const int idx = blockIdx.x * blockDim.x + threadIdx.x;
  if (idx >= NN * dout) return;
  const int ch = idx % dout;
  const float v = X[idx] * scale[ch] + shift[ch];
  X[idx] = v > 0.f ? v : 0.f;
}

__global__ __launch_bounds__(256)
void reparam_kernel(const float* __restrict__ eps, const float* __restrict__ mu,
                    const float* __restrict__ lv, float* __restrict__ v) {
  const int idx = blockIdx.x * blockDim.x + threadIdx.x;
  if (idx >= NN * 64) return;
  v[idx] = eps[idx] * __expf(0.5f * lv[idx]) + mu[idx];
}

// ---------------------------------------------------------------------------
static void run_feast_layer(const float* x, int din, int dout,
                            const float* U, const float* cb, const float* T,
                            const float* bias,
                            const long long* src, const long long* dst,
                            float* Y, float* P, float* agg, const unsigned* deg,
                            float* sums, float* sumsq,
                            float* out, bool with_bn, hipStream_t stream) {
  const int K = 4 * dout;
  hipMemsetAsync(agg, 0, (size_t)NN * dout * sizeof(float), stream);
  if (with_bn) {
    hipMemsetAsync(sums, 0, dout * sizeof(float), stream);
    hipMemsetAsync(sumsq, 0, dout * sizeof(float), stream);
  }
  dim3 gg(NN / 16, K / 16);
  switch (din) {
    case 16: gemm_xT_wmma<16><<<gg, 32, 0, stream>>>(x, T, Y, K); break;
    case 32: gemm_xT_wmma<32><<<gg, 32, 0, stream>>>(x, T, Y, K); break;
    default: gemm_xT_wmma<64><<<gg, 32, 0, stream>>>(x, T, Y, K); break;
  }
  xu_kernel<<<(NN + 255) / 256, 256, 0, stream>>>(x, U, P, din);
  feast_edge_kernel<<<(EE + 255) / 256, 256, 0, stream>>>(src, dst, P, cb, Y, agg, dout);
  if (with_bn)
    feast_finalize_kernel<true><<<512, 256, 0, stream>>>(agg, Y, deg, bias, cb,
                                                         out, sums, sumsq, dout);
  else
    feast_finalize_kernel<false><<<512, 256, 0, stream>>>(agg, Y, deg, bias, cb,
                                                          out, sums, sumsq, dout);
}

extern "C" void kernel_launch(void* const* d_in, const int* in_sizes, int n_in,
                              void* d_out, int out_size, void* d_ws, size_t ws_size,
                              hipStream_t stream) {
  (void)in_sizes; (void)n_in; (void)out_size; (void)ws_size;

  // ---- inputs (setup_inputs dict order, params flattened in insertion order)
  const float*     z   = (const float*)d_in[0];
  const long long* ei  = (const long long*)d_in[1];   // int64 [2, E]
  const float*     eps = (const float*)d_in[2];
  const float *U[6], *cb[6], *T[6], *bb[6];
  for (int li = 0; li < 6; ++li) {
    U[li]  = (const float*)d_in[3 + 4 * li + 0];
    cb[li] = (const float*)d_in[3 + 4 * li + 1];
    T[li]  = (const float*)d_in[3 + 4 * li + 2];
    bb[li] = (const float*)d_in[3 + 4 * li + 3];
  }
  const float *g[4], *beta[4];
  for (int bi = 0; bi < 4; ++bi) {
    g[bi]    = (const float*)d_in[27 + 2 * bi];
    beta[bi] = (const float*)d_in[28 + 2 * bi];
  }
  const long long* src = ei;        // row 0
  const long long* dst = ei + EE;   // row 1

  // ---- workspace carve-up
  uint8_t* w = (uint8_t*)d_ws;
  size_t off = 0;
  float* Y    = (float*)(w + off); off += (size_t)NN * 256 * sizeof(float);
  float* P    = (float*)(w + off); off += (size_t)NN * 4   * sizeof(float);
  float* agg  = (float*)(w + off); off += (size_t)NN * 64  * sizeof(float);
  float* hA   = (float*)(w + off); off += (size_t)NN * 64  * sizeof(float);
  float* hB   = (float*)(w + off); off += (size_t)NN * 64  * sizeof(float);
  unsigned* deg = (unsigned*)(w + off); off += (size_t)NN * sizeof(unsigned);
  float* sums  = (float*)(w + off); off += 64 * sizeof(float);
  float* sumsq = (float*)(w + off); off += 64 * sizeof(float);
  float* scale = (float*)(w + off); off += 64 * sizeof(float);
  float* shift = (float*)(w + off); off += 64 * sizeof(float);

  float* out_v  = (float*)d_out;
  float* out_mu = out_v + (size_t)NN * 64;
  float* out_lv = out_v + (size_t)2 * NN * 64;

  // ---- degrees (shared by all layers)
  hipMemsetAsync(deg, 0, (size_t)NN * sizeof(unsigned), stream);
  deg_count_kernel<<<(EE + 255) / 256, 256, 0, stream>>>(dst, deg);

  // ---- gc7..gc10 with BN + ReLU (ping-pong hA/hB)
  const int dims[4][2] = {{16, 16}, {16, 32}, {32, 32}, {32, 64}};
  const float* cur = z;
  float* bufs[2] = {hA, hB};
  for (int li = 0; li < 4; ++li) {
    const int din = dims[li][0], dout = dims[li][1];
    float* nxt = bufs[li & 1];
    run_feast_layer(cur, din, dout, U[li], cb[li], T[li], bb[li],
                    src, dst, Y, P, agg, deg, sums, sumsq, nxt, true, stream);
    bn_stats_kernel<<<1, 64, 0, stream>>>(sums, sumsq, g[li], beta[li],
                                          scale, shift, dout);
    bn_relu_kernel<<<(NN * dout + 255) / 256, 256, 0, stream>>>(nxt, scale, shift, dout);
    cur = nxt;
  }

  // ---- mu / logvar heads (64 -> 64, no BN), then reparameterize
  run_feast_layer(cur, 64, 64, U[4], cb[4], T[4], bb[4],
                  src, dst, Y, P, agg, deg, sums, sumsq, out_mu, false, stream);
  run_feast_layer(cur, 64, 64, U[5], cb[5], T[5], bb[5],
                  src, dst, Y, P, agg, deg, sums, sumsq, out_lv, false, stream);
  reparam_kernel<<<(NN * 64 + 255) / 256, 256, 0, stream>>>(eps, out_mu, out_lv, out_v);
}